// LinearAttention_57226144252786
// MI455X (gfx1250) — compile-verified
//
#include <hip/hip_runtime.h>

// Linear attention for MI455X (gfx1250, wave32).
// out[n,l,h,v] = phi(Q)[l,:] . KV / (phi(Q)[l,:] . Ksum + eps)
// with KV = phi(K)^T V  (the reference's /S and *S cancel exactly).
//
// HBM-bound (~8 FLOP/byte, 256 MiB traffic -> ~11us floor at 23.3 TB/s):
// stream Q/K/V exactly once, do the 32-wide contractions with
// V_WMMA_F32_16X16X4_F32 (fp32 path: bit-accurate feature map, compute is free).

typedef float v2f __attribute__((ext_vector_type(2)));
typedef float v8f __attribute__((ext_vector_type(8)));

#define N_   8
#define L_   8192
#define S_   8192
#define H_   8
#define D_   32
#define DV_  32
#define NH_  (N_ * H_)
#define RS_  (H_ * D_)      // 256 floats between consecutive l/s rows for fixed (n,h)
#define EPS_ 1e-6f

// phi(x) = elu(x) + 1 = x>0 ? x+1 : exp(x)
__device__ __forceinline__ float elu1(float x) {
    return x > 0.0f ? x + 1.0f : __expf(x);
}

// ---------------------------------------------------------------- zero ws ---
__global__ void la_zero_ws(float* __restrict__ ws, int n) {
    int i = blockIdx.x * blockDim.x + threadIdx.x;
    if (i < n) ws[i] = 0.0f;
}

// -------------------------------------------------- phase 1: KV and Ksum ----
// grid = (NH_, P1_CHUNKS), block = 256 (8 waves). Each wave reduces P1_SPW
// rows of S into a 32x32 KV partial (2x2 WMMA tiles) + Ksum partial. The 8
// waves combine in LDS (ds_add_f32), then the block does ONE atomic flush to
// global workspace (8x fewer device-scope atomics than per-wave flushing).
#define P1_CHUNKS 8
#define P1_WAVES  8
#define P1_SPW    (S_ / (P1_CHUNKS * P1_WAVES))   // 128 s-rows per wave
#define P1_ITERS  (P1_SPW / 4)                    // 32 WMMA steps per wave

__global__ void __launch_bounds__(256)
la_phase1(const float* __restrict__ keys, const float* __restrict__ values,
          float* __restrict__ kv, float* __restrict__ ksum) {
    __shared__ float lds_kv[D_ * DV_];   // 4 KB block-level KV accumulator
    __shared__ float lds_ks[D_];

    const int tid  = threadIdx.x;
    const int nh   = blockIdx.x;
    const int n    = nh / H_, h = nh % H_;
    const int wave = tid >> 5;
    const int lane = tid & 31;
    const int hi   = lane >> 4;          // lane half (ISA A/B layout)
    const int lo   = lane & 15;

    for (int i = tid; i < D_ * DV_; i += 256) lds_kv[i] = 0.0f;
    if (tid < D_) lds_ks[tid] = 0.0f;
    __syncthreads();

    const int s0 = (blockIdx.y * P1_WAVES + wave) * P1_SPW + 2 * hi;

    const float* Kb = keys   + ((size_t)n * S_ * H_ + h) * D_;
    const float* Vb = values + ((size_t)n * S_ * H_ + h) * D_;

    v8f  c00 = {}, c01 = {}, c10 = {}, c11 = {};
    float ks0 = 0.0f, ks1 = 0.0f;

    // Uniform compile-time trip count: keeps the loop a scalar counted loop
    // (no EXEC-mask loop exit near the WMMAs).
#pragma unroll 4
    for (int it = 0; it < P1_ITERS; ++it) {
        // This lane owns rows s0+4*it and s0+4*it+1 of the K/V panels.
        const float* kr0 = Kb + (size_t)(s0 + 4 * it) * RS_;
        const float* kr1 = kr0 + RS_;
        const float* vr0 = Vb + (size_t)(s0 + 4 * it) * RS_;
        const float* vr1 = vr0 + RS_;

        // Prefetch ~4 iterations (16 rows) ahead of the HBM stream.
        __builtin_prefetch(kr0 + 16 * RS_, 0, 0);
        __builtin_prefetch(vr0 + 16 * RS_, 0, 0);

        // A = K^T tiles (m = d, k = s_local), with feature map applied.
        v2f a0, a1, b0, b1;
        a0.x = elu1(kr0[lo]);      a0.y = elu1(kr1[lo]);
        a1.x = elu1(kr0[lo + 16]); a1.y = elu1(kr1[lo + 16]);
        // B = V tiles (k = s_local, n = v).
        b0.x = vr0[lo];            b0.y = vr1[lo];
        b1.x = vr0[lo + 16];       b1.y = vr1[lo + 16];

        // Ksum partials ride on the already-ELU'd A operands.
        ks0 += a0.x + a0.y;
        ks1 += a1.x + a1.y;

        c00 = __builtin_amdgcn_wmma_f32_16x16x4_f32(false, a0, false, b0, (short)0, c00, false, false);
        c01 = __builtin_amdgcn_wmma_f32_16x16x4_f32(false, a0, false, b1, (short)0, c01, false, false);
        c10 = __builtin_amdgcn_wmma_f32_16x16x4_f32(false, a1, false, b0, (short)0, c10, false, false);
        c11 = __builtin_amdgcn_wmma_f32_16x16x4_f32(false, a1, false, b1, (short)0, c11, false, false);
    }

    // Ksum[d=lo(+16)]: lane halves hold disjoint s subsets of the same d.
    ks0 += __shfl_xor(ks0, 16, 32);
    ks1 += __shfl_xor(ks1, 16, 32);
    if (lane < 16) {
        atomicAdd(&lds_ks[lo],      ks0);
        atomicAdd(&lds_ks[lo + 16], ks1);
    }

    // C/D layout: VGPR r, lane -> element (m = r + 8*hi, n = lo). Combine the
    // 8 waves' partials in LDS (ds_add_f32).
#pragma unroll
    for (int r = 0; r < 8; ++r) {
        const int m = r + 8 * hi;
        atomicAdd(&lds_kv[(m)      * DV_ + lo],      c00[r]);
        atomicAdd(&lds_kv[(m)      * DV_ + lo + 16], c01[r]);
        atomicAdd(&lds_kv[(m + 16) * DV_ + lo],      c10[r]);
        atomicAdd(&lds_kv[(m + 16) * DV_ + lo + 16], c11[r]);
    }
    __syncthreads();

    // One cooperative flush per block to the global accumulators.
    float* kvb = kv + nh * (D_ * DV_);
    for (int i = tid; i < D_ * DV_; i += 256) atomicAdd(&kvb[i], lds_kv[i]);
    if (tid < D_) atomicAdd(&ksum[nh * 32 + tid], lds_ks[tid]);
}

// ------------------------------------- phase 2: out = phi(Q) @ KV * Z ------
// grid.x = NH_ * (L_/(16*8)), block = 256. Each wave produces one 16-row
// l-tile: 8 WMMA steps over D=32, 2 v-tiles, plus the Z normalizer.
// KV/Ksum are 4KB+128B per head -> L2-resident (192 MB L2), so re-reading
// them per wave is free; Q and out stream once through HBM.
#define P2_TPB (L_ / (16 * P1_WAVES))   // 64 blocks per (n,h)

__global__ void __launch_bounds__(256)
la_phase2(const float* __restrict__ queries, const float* __restrict__ kv,
          const float* __restrict__ ksum, float* __restrict__ out) {
    const int wave = threadIdx.x >> 5;
    const int lane = threadIdx.x & 31;
    const int hi   = lane >> 4;
    const int lo   = lane & 15;

    const int nh = blockIdx.x / P2_TPB;
    const int l0 = ((blockIdx.x % P2_TPB) * P1_WAVES + wave) * 16;
    const int n  = nh / H_, h = nh % H_;

    const float* Qb  = queries + ((size_t)n * L_ * H_ + h) * D_;
    const float* kvb = kv   + nh * (D_ * DV_);
    const float* ksb = ksum + nh * 32;

    v8f   c0 = {}, c1 = {};
    float z  = 0.0f;

#pragma unroll
    for (int d = 0; d < D_; d += 4) {
        // A = Q tile (m = l row, k = d_local): lane reads Q[l0+lo][d+2*hi+j].
        const float* q = Qb + (size_t)(l0 + lo) * RS_ + d + 2 * hi;
        v2f a;
        a.x = elu1(q[0]);
        a.y = elu1(q[1]);
        // Z partial: this lane covers d+2*hi{,+1} for row lo.
        z += a.x * ksb[d + 2 * hi] + a.y * ksb[d + 2 * hi + 1];
        // B = KV tile (k = d_local, n = v).
        v2f b0, b1;
        b0.x = kvb[(d + 2 * hi) * DV_ + lo];          b0.y = kvb[(d + 2 * hi + 1) * DV_ + lo];
        b1.x = kvb[(d + 2 * hi) * DV_ + lo + 16];     b1.y = kvb[(d + 2 * hi + 1) * DV_ + lo + 16];

        c0 = __builtin_amdgcn_wmma_f32_16x16x4_f32(false, a, false, b0, (short)0, c0, false, false);
        c1 = __builtin_amdgcn_wmma_f32_16x16x4_f32(false, a, false, b1, (short)0, c1, false, false);
    }

    // Full row-lo dot(phi(Q), Ksum) = lane-half partial + opposite half.
    z += __shfl_xor(z, 16, 32);
    const float zinv = 1.0f / (z + EPS_);   // all lanes: zinv for row `lo`

    float* Ob = out + ((size_t)n * L_ * H_ + h) * DV_;
#pragma unroll
    for (int r = 0; r < 8; ++r) {
        const int   m  = r + 8 * hi;                 // output row within tile
        const float zi = __shfl(zinv, m, 32);        // broadcast row-m normalizer
        float* orow = Ob + (size_t)(l0 + m) * RS_;
        orow[lo]      = c0[r] * zi;
        orow[lo + 16] = c1[r] * zi;
    }
}

// ---------------------------------------------------------------------------
extern "C" void kernel_launch(void* const* d_in, const int* in_sizes, int n_in,
                              void* d_out, int out_size, void* d_ws, size_t ws_size,
                              hipStream_t stream) {
    (void)in_sizes; (void)n_in; (void)out_size; (void)ws_size;
    const float* q = (const float*)d_in[0];
    const float* k = (const float*)d_in[1];
    const float* v = (const float*)d_in[2];
    float* out  = (float*)d_out;
    float* ws   = (float*)d_ws;
    float* kv   = ws;                       // NH_ * 32 * 32 = 65536 floats
    float* ksum = ws + NH_ * D_ * DV_;      // NH_ * 32     =  2048 floats

    const int zn = NH_ * D_ * DV_ + NH_ * D_;
    la_zero_ws<<<(zn + 255) / 256, 256, 0, stream>>>(ws, zn);
    la_phase1<<<dim3(NH_, P1_CHUNKS), 256, 0, stream>>>(k, v, kv, ksum);
    la_phase2<<<dim3(NH_ * P2_TPB), 256, 0, stream>>>(q, kv, ksum, out);
}